// PCENLayer_44461501448517
// MI455X (gfx1250) — compile-verified
//
#include <hip/hip_runtime.h>
#include <hip/hip_bf16.h>
#include <math.h>

typedef __attribute__((ext_vector_type(2))) float v2f;
typedef __attribute__((ext_vector_type(4))) float v4f;
typedef __attribute__((ext_vector_type(8))) float v8f;

#define KLEN   128        // FIR taps
#define TILE   256        // outputs per WMMA tile (16 blocks x 16 times)
#define WIN    384        // window floats per tile (KLEN halo + TILE)
#define NWAVES 8
#define TT     16384
#define NCH    64
#define NB     16
#define NTILES (TT / TILE)   // 64
#define PCEN_EPS 1e-6f

__global__ __launch_bounds__(256) void pcen_wmma_kernel(
    const float* __restrict__ x,
    const float* __restrict__ log_alpha,
    const float* __restrict__ log_delta,
    const float* __restrict__ log_root,
    const float* __restrict__ log_s,
    float* __restrict__ out)
{
    __shared__ __align__(16) float sW[KLEN];            // normalized EMA impulse
    __shared__ float sSum;
    __shared__ __align__(16) float sWin[NWAVES][WIN];   // per-wave |x| window
    __shared__ __align__(16) float sTr[NWAVES][TILE];   // per-wave transpose buffer

    const int c = blockIdx.x % NCH;
    const int b = blockIdx.x / NCH;
    const long rowBase = ((long)b * NCH + c) * (long)TT;

    const int tid  = threadIdx.x;
    const int lane = tid & 31;
    const int wav  = tid >> 5;
    const int i    = lane & 15;   // M-row within half
    const int half = lane >> 4;   // 0: lanes 0-15, 1: lanes 16-31
    const int n    = i;           // N (column) index for B/C/D layout

    // ---- per-channel scalars (redundant per thread; trivial cost) ----
    const float s     = 1.0f / (1.0f + __expf(-log_s[c]));
    const float alpha = 1.0f / (1.0f + __expf(-log_alpha[c]));
    const float delta = log1pf(__expf(log_delta[c])) + 1e-6f;   // softplus
    const float root  = log1pf(__expf(log_root[c])) + 0.1f;
    const float droot = __powf(delta, root);
    const float nalpha = -alpha;  // fold 1/gain into the exponent (kills the divide)

    // ---- EMA impulse: imp[j] = s*(1-s)^j, normalized by (sum + 1e-8) ----
    if (tid < KLEN) sW[tid] = s * __powf(1.0f - s, (float)tid);
    __syncthreads();
    if (tid == 0) {
        float acc = 0.0f;
        for (int j = 0; j < KLEN; ++j) acc += sW[j];
        sSum = acc + 1e-8f;
    }
    __syncthreads();
    if (tid < KLEN) sW[tid] = sW[tid] / sSum;
    __syncthreads();

    // ---- build Toeplitz weight operands (A matrices), register-resident ----
    // W_a[i][j] = w[16a + i - j]; A chunk (a, j0): A[i][k] = W_a[i][j0+k], k=0..3
    // f32 16x4 A layout: VGPR0 = K row (j0 + 2*half), VGPR1 = (j0 + 2*half + 1)
    v2f aOp[36];
#pragma unroll
    for (int ins = 0; ins < 36; ++ins) {
        const int ablk = ins >> 2;
        const int j0   = (ins & 3) << 2;
        const int q    = 16 * ablk + i - j0 - 2 * half;   // tap index for VGPR0
        float a0 = (q     >= 0 && q     < KLEN) ? sW[q]     : 0.0f;
        float a1 = (q - 1 >= 0 && q - 1 < KLEN) ? sW[q - 1] : 0.0f;
        aOp[ins][0] = a0;
        aOp[ins][1] = a1;
    }

    float* win = sWin[wav];
    float* tr  = sTr[wav];

    // each wave owns tiles {wav, wav+8, ..., wav+56} of this row's 64 tiles
    for (int tb = 0; tb < NTILES; tb += NWAVES) {
        const int tile = tb + wav;
        const int t0   = tile * TILE;
        const float* src = x + rowBase + (t0 - KLEN);

        // ---- load |x| window [t0-128, t0+256) into this wave's LDS slice ----
        if (tile == 0) {
            // slow path: left zero-pad (only wave 0, first iteration)
#pragma unroll
            for (int rseg = 0; rseg < WIN / 32; ++rseg) {
                const int pos = rseg * 32 + lane;
                const int t   = t0 - KLEN + pos;
                float v = 0.0f;
                if (t >= 0) v = fabsf(x[rowBase + t]);
                win[pos] = v;
            }
        } else {
            // fast path: unguarded coalesced loads (scalar branch; no cndmask)
#pragma unroll
            for (int rseg = 0; rseg < WIN / 32; ++rseg) {
                const int pos = rseg * 32 + lane;
                win[pos] = fabsf(src[pos]);
            }
        }
        // prefetch next window for this wave (global_prefetch_b8 path)
        if (tile + NWAVES < NTILES) {
            __builtin_prefetch(&x[rowBase + (t0 + NWAVES * TILE - KLEN) + lane * 4], 0, 0);
        }
        __syncthreads();

        // ---- block-FIR via chained f32 WMMA: D = sum_a W_a x X_a ----
        v8f acc = {};
#pragma unroll
        for (int ins = 0; ins < 36; ++ins) {
            const int ablk = ins >> 2;
            const int j0   = (ins & 3) << 2;
            // B operand (v2f): rows {j0+2*half, j0+2*half+1}, column n
            const int off  = (KLEN - 16 * ablk) + 16 * n + j0 + 2 * half;  // even -> 8B aligned
            const v2f bOp  = *(const v2f*)&win[off];
            acc = __builtin_amdgcn_wmma_f32_16x16x4_f32(
                      false, aOp[ins], false, bOp, (short)0, acc, false, false);
        }

        // ---- PCEN epilogue; stage to LDS for coalesced stores ----
        // C/D layout: VGPR r -> M = r + 8*half, N = lane%16; t = t0 + 16N + M
#pragma unroll
        for (int r = 0; r < 8; ++r) {
            const int mrow = r + 8 * half;
            const int tl   = 16 * n + mrow;
            const float M  = acc[r];
            const float xm = win[KLEN + tl];
            const float invg   = __powf(PCEN_EPS + M, nalpha);  // 1/gain, no divide
            const float normed = xm * invg;
            tr[tl] = __powf(normed + delta, root) - droot;
        }
        __syncthreads();

        // ---- coalesced b128 stores ----
#pragma unroll
        for (int part = 0; part < 2; ++part) {
            const int o = part * 128 + lane * 4;
            const v4f v = *(const v4f*)&tr[o];
            *(v4f*)&out[rowBase + t0 + o] = v;
        }
        __syncthreads();
    }
}

extern "C" void kernel_launch(void* const* d_in, const int* in_sizes, int n_in,
                              void* d_out, int out_size, void* d_ws, size_t ws_size,
                              hipStream_t stream) {
    const float* x         = (const float*)d_in[0];
    const float* log_alpha = (const float*)d_in[1];
    const float* log_delta = (const float*)d_in[2];
    const float* log_root  = (const float*)d_in[3];
    const float* log_s     = (const float*)d_in[4];
    float* out = (float*)d_out;

    dim3 grid(NB * NCH);   // one block per (b, c) row
    dim3 block(256);       // 8 waves x 32 lanes
    pcen_wmma_kernel<<<grid, block, 0, stream>>>(x, log_alpha, log_delta,
                                                 log_root, log_s, out);
}